// CombinedMarginLoss_30039001268428
// MI455X (gfx1250) — compile-verified
//
#include <hip/hip_runtime.h>

// ---------------------------------------------------------------------------
// CombinedMarginLoss forward (ArcFace m1=1, m2=0.5, m3=0 + interclass filter)
//   out[i,j] = 64 * ( j==label[i] ? margin(x) : (x > 0.3 ? 0 : x) )
//
// Two-kernel split:
//   1) margin_stream_kernel: memory-bound streamer, 3 VALU/element
//      (820 MB traffic -> ~35us floor at 23.3 TB/s; VALU work fully hidden).
//      gfx1250 path: global_load_async_to_lds_b128 double-buffered pipeline
//      (ASYNCcnt) + non-temporal 128-bit stores. Full 256-lane stages:
//      25.6M float4 total is divisible by 1024 v4/block -> no tail, no
//      divergence, no exec-mask overhead.
//   2) margin_fixup_kernel: one lane per row applies the sqrt-based margin to
//      the target logit and overwrites out[row,label]. Negligible cost.
// No matmul structure in this op -> WMMA is intentionally not used.
// ---------------------------------------------------------------------------

typedef __attribute__((ext_vector_type(4))) float v4f;
typedef int v4i __attribute__((vector_size(16)));   // matches builtin proto

#define TPB       256
#define STAGE_V4  256                    // one float4 per lane per stage (4 KB)
#define NSTAGES   4
#define SEG_V4    (STAGE_V4 * NSTAGES)   // 1024 float4 per block
#define C_DIM     100000

#define K_SCALE   64.0f
#define K_THRESH  0.3f
#define K_COS_M   0.87758256189037271612f
#define K_SIN_M   0.47942553860420300027f
#define K_THETA  (-0.87758256189037271612f)   // cos(pi - 0.5)
#define K_SINMM   0.23971276930210150013f     // sin(pi - 0.5) * 0.5

// ---- gfx1250 async global->LDS path (guarded; falls back to direct loads) --
#if defined(__gfx1250__) && __has_builtin(__builtin_amdgcn_global_load_async_to_lds_b128)
#define USE_ASYNC 1
#else
#define USE_ASYNC 0
#endif

#if defined(__gfx1250__)
#if __has_builtin(__builtin_amdgcn_s_wait_asynccnt)
#define WAIT_ASYNC(n) __builtin_amdgcn_s_wait_asynccnt(n)
#else
#define WAIT_ASYNC(n) asm volatile("s_wait_asynccnt %0" ::"i"(n) : "memory")
#endif
#else
#define WAIT_ASYNC(n)
#endif

#if USE_ASYNC
// arg0: int4 addrspace(1)* (global src), arg1: int4 addrspace(3)* (LDS dst).
// Integer round-trip is value-correct: AS1 == low 64b of generic global addr;
// AS3 == low 32b of generic LDS addr (flat aperture rule), AS3 ptrs are 32-bit.
#define GPTR(p) ((__attribute__((address_space(1))) v4i*)(unsigned long long)(uintptr_t)(p))
#define LPTR(p) ((__attribute__((address_space(3))) v4i*)(unsigned int)(uintptr_t)(p))
#endif

// Bulk path: interclass filter + scale only (target column fixed up later).
__device__ __forceinline__ float filt(float x) {
  return (x > K_THRESH) ? 0.0f : x * K_SCALE;   // cmp + mul + cndmask
}

__device__ __forceinline__ v4f filt4(v4f x) {
  v4f r;
  r.x = filt(x.x);
  r.y = filt(x.y);
  r.z = filt(x.z);
  r.w = filt(x.w);
  return r;
}

__global__ __launch_bounds__(TPB) void margin_stream_kernel(
    const float* __restrict__ logits, float* __restrict__ out) {
  const int tid = threadIdx.x;
  const size_t base_v4 = (size_t)blockIdx.x * SEG_V4;   // 1D over whole buffer
  const v4f* __restrict__ gsrc = (const v4f*)logits + base_v4;
  v4f* __restrict__ gdst       = (v4f*)out + base_v4;

#if USE_ASYNC
  __shared__ v4f buf[2][STAGE_V4];                      // 8 KB double buffer

  __builtin_amdgcn_global_load_async_to_lds_b128(
      GPTR(gsrc + tid), LPTR(&buf[0][tid]), 0, 0);
#pragma unroll
  for (int s = 0; s < NSTAGES; ++s) {
    if (s + 1 < NSTAGES) {                              // compile-time uniform
      __builtin_amdgcn_global_load_async_to_lds_b128(
          GPTR(gsrc + (s + 1) * STAGE_V4 + tid),
          LPTR(&buf[(s + 1) & 1][tid]), 0, 0);
      WAIT_ASYNC(1);                                    // stage s landed (in-order)
    } else {
      WAIT_ASYNC(0);                                    // last stage landed
    }
    v4f x = buf[s & 1][tid];                            // own lane's 16B; no barrier
    __builtin_nontemporal_store(filt4(x), gdst + s * STAGE_V4 + tid);
  }
#else
  // Fallback: direct 128-bit non-temporal streaming (still near roofline).
#pragma unroll
  for (int s = 0; s < NSTAGES; ++s) {
    v4f x = __builtin_nontemporal_load(gsrc + s * STAGE_V4 + tid);
    __builtin_nontemporal_store(filt4(x), gdst + s * STAGE_V4 + tid);
  }
#endif
}

// Scalar tail (only launched if total elements aren't a multiple of the tile).
__global__ __launch_bounds__(TPB) void margin_tail_kernel(
    const float* __restrict__ logits, float* __restrict__ out,
    long long start, long long total) {
  const long long i = start + (long long)blockIdx.x * TPB + threadIdx.x;
  if (i < total) out[i] = filt(logits[i]);
}

// One lane per row: ArcFace additive angular margin on the target logit.
__global__ __launch_bounds__(TPB) void margin_fixup_kernel(
    const float* __restrict__ logits, const int* __restrict__ labels,
    float* __restrict__ out, int B) {
  const int r = blockIdx.x * TPB + threadIdx.x;
  if (r < B) {
    const int lab = labels[r];
    const size_t idx = (size_t)r * C_DIM + (size_t)lab;
    const float x = logits[idx];                        // original (unfiltered)
    const float s  = __builtin_sqrtf(__builtin_fmaxf(1.0f - x * x, 0.0f));
    const float cm = x * K_COS_M - s * K_SIN_M;
    const float v  = (x > K_THETA) ? cm : (x - K_SINMM);
    out[idx] = v * K_SCALE;
  }
}

extern "C" void kernel_launch(void* const* d_in, const int* in_sizes, int n_in,
                              void* d_out, int out_size, void* d_ws, size_t ws_size,
                              hipStream_t stream) {
  (void)n_in; (void)d_ws; (void)ws_size;
  const float* logits = (const float*)d_in[0];   // (B, 100000) fp32
  const int*   labels = (const int*)d_in[1];     // (B,) int (harness contract)
  float*       out    = (float*)d_out;           // (B, 100000) fp32
  const int B = in_sizes[1];                     // 1024 rows

  const long long total    = (long long)out_size;       // 102,400,000 elements
  const long long total_v4 = total / 4;                 // 25,600,000 (exact)
  const long long nblocks  = total_v4 / SEG_V4;         // 25,000 for B=1024
  const long long done     = nblocks * SEG_V4 * 4;      // elements covered

  // 1) bulk stream: double-buffered async global->LDS pipeline
  if (nblocks > 0)
    margin_stream_kernel<<<dim3((unsigned)nblocks, 1, 1), dim3(TPB, 1, 1), 0,
                           stream>>>(logits, out);
  // 1b) tail (zero blocks for this problem size; kept for generality)
  if (done < total) {
    const long long rem = total - done;
    margin_tail_kernel<<<dim3((unsigned)((rem + TPB - 1) / TPB), 1, 1),
                         dim3(TPB, 1, 1), 0, stream>>>(logits, out, done, total);
  }
  // 2) per-row target-logit margin fixup (ordered after on same stream)
  margin_fixup_kernel<<<dim3((B + TPB - 1) / TPB, 1, 1), dim3(TPB, 1, 1), 0,
                        stream>>>(logits, labels, out, B);
}